// Encoder_49572512530717
// MI455X (gfx1250) — compile-verified
//
#include <hip/hip_runtime.h>
#include <math.h>

typedef __attribute__((ext_vector_type(2))) float v2f;
typedef __attribute__((ext_vector_type(8))) float v8f;

#define C128 128
#define SA_STRIDE 130            // padded A row stride (floats) -> conflict-free b64 reads
#define SW_STRIDE 264            // padded W k-pair stride (floats): 128 cols * 2 + 8 pad

// ---------------------------------------------------------------------------
// WMMA GEMM: Out[M,128] = A[M,128] @ W[128,128] (+bias) (+tanh)
// mode: 0 = raw, 1 = +bias, 2 = +bias then tanh
// Block = 256 threads (8 wave32). Block covers 128 rows; wave w owns rows
// [w*16, w*16+16) of the block strip and all 128 output columns (8 tiles).
//
// LDS layouts:
//   sA[r * SA_STRIDE + k]                      : A strip, row-major, padded
//   sW[(k/2) * SW_STRIDE + j*2 + (k&1)]        : W k-pair interleaved, so the
//     B fragment {W[k][j], W[k+1][j]} is one aligned 8-byte ds_load_b64.
// ---------------------------------------------------------------------------
__global__ void __launch_bounds__(256)
gemm128_wmma(const float* __restrict__ A, const float* __restrict__ W,
             const float* __restrict__ bias, float* __restrict__ Out,
             int M, int mode) {
  __shared__ float sW[64 * SW_STRIDE];        // ~67.6 KB
  __shared__ float sA[C128 * SA_STRIDE];      // ~66.6 KB
  const int tid  = threadIdx.x;
  const int wave = tid >> 5;
  const int lane = tid & 31;
  const int row0 = blockIdx.x * C128;

  // Stage W (row-major [k][j] in global) into k-pair interleaved LDS layout.
  for (int i = tid * 4; i < C128 * C128; i += 256 * 4) {
    float4 v = *(const float4*)&W[i];
    int k = i >> 7, j = i & 127;               // j, j+1, j+2, j+3
    float* dstp = &sW[(k >> 1) * SW_STRIDE + (k & 1)];
    dstp[(j + 0) * 2] = v.x;
    dstp[(j + 1) * 2] = v.y;
    dstp[(j + 2) * 2] = v.z;
    dstp[(j + 3) * 2] = v.w;
  }
  // Stage A strip (padded rows).
  for (int i = tid * 4; i < C128 * C128; i += 256 * 4) {
    int r = i >> 7, c = i & 127;
    int gr = row0 + r;
    float4 v = make_float4(0.f, 0.f, 0.f, 0.f);
    if (gr < M) v = *(const float4*)&A[(size_t)gr * C128 + c];
    float* dstp = &sA[r * SA_STRIDE + c];
    dstp[0] = v.x; dstp[1] = v.y; dstp[2] = v.z; dstp[3] = v.w;
  }
  __syncthreads();

  // A fragment (16x4 f32): lane l -> m = l&15, k = 2*(l>>4) + {0,1}
  const int arow = wave * 16 + (lane & 15);
  const int koff = (lane >> 4) * 2;            // 0 or 2
  const int bn   = lane & 15;
  const float* __restrict__ aBase = &sA[arow * SA_STRIDE + koff];
  const float* __restrict__ bBase = &sW[(koff >> 1) * SW_STRIDE + bn * 2];

  v8f acc[8] = {};
  for (int k = 0; k < C128; k += 4) {
    v2f a = *(const v2f*)(aBase + k);                       // ds_load_b64
    const float* bk = bBase + (k >> 1) * SW_STRIDE;
#pragma unroll
    for (int t = 0; t < 8; ++t) {
      v2f b = *(const v2f*)(bk + t * 32);                   // ds_load_b64
      acc[t] = __builtin_amdgcn_wmma_f32_16x16x4_f32(
          false, a, false, b, (short)0, acc[t], false, false);
    }
  }

  // C/D layout: VGPR v, lane l -> m = v + 8*(l>>4), n = l&15
  const int mBase = wave * 16 + ((lane >> 4) ? 8 : 0);
#pragma unroll
  for (int t = 0; t < 8; ++t) {
#pragma unroll
    for (int v = 0; v < 8; ++v) {
      int gr = row0 + mBase + v;
      if (gr < M) {
        int col  = t * 16 + bn;
        float val = acc[t][v];
        if (mode >= 1) val += bias[col];
        if (mode == 2) val = tanhf(val);
        Out[(size_t)gr * C128 + col] = val;
      }
    }
  }
}

// ---------------------------------------------------------------------------
// Elementwise / segment kernels
// ---------------------------------------------------------------------------
__device__ __forceinline__ void atomicMaxFloat(float* addr, float val) {
  int iv = __float_as_int(val);
  if (iv >= 0) atomicMax((int*)addr, iv);
  else         atomicMin((unsigned int*)addr, (unsigned int)iv);
}

__global__ void k_fill(float* p, float v, int n) {
  int i = blockIdx.x * 256 + threadIdx.x;
  if (i < n) p[i] = v;
}

// deg[dst] += 1 ; loop_sum[dst][c] += eattr[e][c]     (thread per (e,c))
__global__ void k_edge_accum(const int* __restrict__ dst,
                             const float* __restrict__ eattr,
                             float* __restrict__ loop_sum,
                             float* __restrict__ deg, int E) {
  int i = blockIdx.x * 256 + threadIdx.x;
  if (i >= E * C128) return;
  int e = i >> 7, c = i & 127;
  int d = dst[e];
  atomicAdd(&loop_sum[(size_t)d * C128 + c], eattr[i]);
  if (c == 0) atomicAdd(&deg[d], 1.0f);
}

// loop_attr = loop_sum / max(deg, 1)
__global__ void k_loop_fin(float* __restrict__ loopA,
                           const float* __restrict__ deg, int n) {
  int i = blockIdx.x * 256 + threadIdx.x;
  if (i >= n) return;
  float d = deg[i >> 7];
  loopA[i] = loopA[i] / (d > 1.0f ? d : 1.0f);
}

// warp per edge: logit = dot(leaky_relu(xl[s]+xr[d]+el, 0.2), att); seg-max
__global__ void k_logits(const int* __restrict__ src, const int* __restrict__ dst,
                         const float* __restrict__ xl, const float* __restrict__ xr,
                         const float* __restrict__ el_e, const float* __restrict__ el_n,
                         const float* __restrict__ att,
                         float* __restrict__ logits, float* __restrict__ maxb,
                         int E, int N) {
  int w = blockIdx.x * 8 + (threadIdx.x >> 5);
  int lane = threadIdx.x & 31;
  if (w >= E + N) return;
  int s, d; const float* el;
  if (w < E) { s = src[w]; d = dst[w]; el = el_e + (size_t)w * C128; }
  else       { s = d = w - E;          el = el_n + (size_t)(w - E) * C128; }
  float acc = 0.f;
#pragma unroll
  for (int j = 0; j < 4; ++j) {
    int c = lane + j * 32;
    float h = xl[(size_t)s * C128 + c] + xr[(size_t)d * C128 + c] + el[c];
    h = h > 0.f ? h : 0.2f * h;
    acc += h * att[c];
  }
  for (int off = 16; off > 0; off >>= 1) acc += __shfl_xor(acc, off, 32);
  if (lane == 0) {
    logits[w] = acc;
    atomicMaxFloat(&maxb[d], acc);
  }
}

// expbuf = exp(logit - max[dst]); sum[dst] += expbuf
__global__ void k_expsum(const int* __restrict__ dst,
                         const float* __restrict__ logits,
                         const float* __restrict__ maxb,
                         float* __restrict__ expb, float* __restrict__ sumb,
                         int E, int N) {
  int e = blockIdx.x * 256 + threadIdx.x;
  if (e >= E + N) return;
  int d = (e < E) ? dst[e] : e - E;
  float ex = expf(logits[e] - maxb[d]);
  expb[e] = ex;
  atomicAdd(&sumb[d], ex);
}

// out[dst][c] += (exp/sum[dst]) * xl[src][c]      (thread per (e,c))
__global__ void k_scatter(const int* __restrict__ src, const int* __restrict__ dst,
                          const float* __restrict__ xl,
                          const float* __restrict__ expb, const float* __restrict__ sumb,
                          float* __restrict__ out, int E, int N) {
  int i = blockIdx.x * 256 + threadIdx.x;
  if (i >= (E + N) * C128) return;
  int e = i >> 7, c = i & 127;
  int s, d;
  if (e < E) { s = src[e]; d = dst[e]; }
  else       { s = d = e - E; }
  float alpha = expb[e] / sumb[d];
  atomicAdd(&out[(size_t)d * C128 + c], alpha * xl[(size_t)s * C128 + c]);
}

__global__ void k_bias_bn_tanh(float* __restrict__ h, const float* __restrict__ bias,
                               const float* __restrict__ g, const float* __restrict__ b,
                               const float* __restrict__ m, const float* __restrict__ v,
                               int n) {
  int i = blockIdx.x * 256 + threadIdx.x;
  if (i >= n) return;
  int c = i & 127;
  float x = h[i] + bias[c];
  x = (x - m[c]) * rsqrtf(v[c] + 1e-5f) * g[c] + b[c];
  h[i] = tanhf(x);
}

__global__ void k_bn_tanh(float* __restrict__ t,
                          const float* __restrict__ g, const float* __restrict__ b,
                          const float* __restrict__ m, const float* __restrict__ v,
                          int n) {
  int i = blockIdx.x * 256 + threadIdx.x;
  if (i >= n) return;
  int c = i & 127;
  float x = (t[i] - m[c]) * rsqrtf(v[c] + 1e-5f) * g[c] + b[c];
  t[i] = tanhf(x);
}

__global__ void k_seg_max(const float* __restrict__ gate, const int* __restrict__ batch,
                          float* __restrict__ gmax, int n) {
  int i = blockIdx.x * 256 + threadIdx.x;
  if (i >= n) return;
  atomicMaxFloat(&gmax[(size_t)batch[i >> 7] * C128 + (i & 127)], gate[i]);
}

__global__ void k_seg_exp(float* __restrict__ gate, const int* __restrict__ batch,
                          const float* __restrict__ gmax, float* __restrict__ gsum, int n) {
  int i = blockIdx.x * 256 + threadIdx.x;
  if (i >= n) return;
  size_t gi = (size_t)batch[i >> 7] * C128 + (i & 127);
  float e = expf(gate[i] - gmax[gi]);
  gate[i] = e;
  atomicAdd(&gsum[gi], e);
}

__global__ void k_seg_pool(const float* __restrict__ gate, const int* __restrict__ batch,
                           const float* __restrict__ gsum, const float* __restrict__ h,
                           float* __restrict__ pooled, int n) {
  int i = blockIdx.x * 256 + threadIdx.x;
  if (i >= n) return;
  size_t gi = (size_t)batch[i >> 7] * C128 + (i & 127);
  atomicAdd(&pooled[gi], (gate[i] / gsum[gi]) * h[i]);
}

// warp per row: out = row / max(||row||, 1e-12)
__global__ void k_l2norm(const float* __restrict__ in, float* __restrict__ out, int M) {
  int row = blockIdx.x * 8 + (threadIdx.x >> 5);
  int lane = threadIdx.x & 31;
  if (row >= M) return;
  float vals[4], ss = 0.f;
#pragma unroll
  for (int j = 0; j < 4; ++j) {
    vals[j] = in[(size_t)row * C128 + lane + j * 32];
    ss += vals[j] * vals[j];
  }
  for (int off = 16; off > 0; off >>= 1) ss += __shfl_xor(ss, off, 32);
  float norm = sqrtf(ss);
  float inv = 1.f / (norm < 1e-12f ? 1e-12f : norm);
#pragma unroll
  for (int j = 0; j < 4; ++j) out[(size_t)row * C128 + lane + j * 32] = vals[j] * inv;
}

// ---------------------------------------------------------------------------
// Host orchestration
// ---------------------------------------------------------------------------
extern "C" void kernel_launch(void* const* d_in, const int* in_sizes, int n_in,
                              void* d_out, int out_size, void* d_ws, size_t ws_size,
                              hipStream_t stream) {
  const float* x     = (const float*)d_in[0];
  const int*   eidx  = (const int*)d_in[1];
  const float* eattr = (const float*)d_in[2];
  const int*   batch = (const int*)d_in[3];
  const float* imgf  = (const float*)d_in[4];
  const float* txtf  = (const float*)d_in[5];
  const int N = in_sizes[0] / C128;
  const int E = in_sizes[1] / 2;
  const int G = in_sizes[4] / C128;
  const int* src = eidx;
  const int* dst = eidx + E;

  // params in setup_inputs() insertion order
  int p = 6;
  const float* c1Wl = (const float*)d_in[p++];
  const float* c1Wr = (const float*)d_in[p++];
  const float* c1We = (const float*)d_in[p++];
  const float* c1at = (const float*)d_in[p++];
  const float* c1bi = (const float*)d_in[p++];
  const float* c2Wl = (const float*)d_in[p++];
  const float* c2Wr = (const float*)d_in[p++];
  const float* c2We = (const float*)d_in[p++];
  const float* c2at = (const float*)d_in[p++];
  const float* c2bi = (const float*)d_in[p++];
  const float* bn1g = (const float*)d_in[p++]; const float* bn1b = (const float*)d_in[p++];
  const float* bn1m = (const float*)d_in[p++]; const float* bn1v = (const float*)d_in[p++];
  const float* bn2g = (const float*)d_in[p++]; const float* bn2b = (const float*)d_in[p++];
  const float* bn2m = (const float*)d_in[p++]; const float* bn2v = (const float*)d_in[p++];
  const float* agW1 = (const float*)d_in[p++]; const float* agb1 = (const float*)d_in[p++];
  const float* agW2 = (const float*)d_in[p++]; const float* agb2 = (const float*)d_in[p++];
  const float* liW  = (const float*)d_in[p++]; const float* lib  = (const float*)d_in[p++];
  const float* imW  = (const float*)d_in[p++]; const float* imb  = (const float*)d_in[p++];
  const float* big  = (const float*)d_in[p++]; const float* bib  = (const float*)d_in[p++];
  const float* bim  = (const float*)d_in[p++]; const float* biv  = (const float*)d_in[p++];
  const float* txW  = (const float*)d_in[p++]; const float* txb  = (const float*)d_in[p++];
  const float* btg  = (const float*)d_in[p++]; const float* btb  = (const float*)d_in[p++];
  const float* btm  = (const float*)d_in[p++]; const float* btv  = (const float*)d_in[p++];

  // workspace layout (floats, all 16B aligned)
  const size_t NC = (size_t)N * C128;
  const size_t EC = (size_t)E * C128;
  const size_t E2 = (size_t)E + N;
  float* w = (float*)d_ws;
  float* el_e   = w; w += EC;
  float* xl     = w; w += NC;
  float* xr     = w; w += NC;
  float* el_n   = w; w += NC;
  float* loopA  = w; w += NC;
  float* h1     = w; w += NC;
  float* h2     = w; w += NC;
  float* logits = w; w += E2;
  float* expb   = w; w += E2;
  float* deg    = w; w += N;
  float* maxb   = w; w += N;
  float* sumb   = w; w += N;
  float* gmax   = w; w += (size_t)G * C128;
  float* gsum   = w; w += (size_t)G * C128;
  float* pooled = w; w += (size_t)G * C128;
  float* tmp1   = w; w += (size_t)G * C128;
  float* tmp2   = w; w += (size_t)G * C128;
  float* tmp3   = w; w += (size_t)G * C128;

  auto gemm = [&](const float* A, const float* Wm, const float* bias, float* Out,
                  int M, int mode) {
    gemm128_wmma<<<(M + 127) / 128, 256, 0, stream>>>(A, Wm, bias, Out, M, mode);
  };
  const float NEG_INF = -__builtin_inff();

  // Self-loop edge attrs (mean of incoming), computed once
  hipMemsetAsync(deg, 0, (size_t)N * sizeof(float), stream);
  hipMemsetAsync(loopA, 0, NC * sizeof(float), stream);
  k_edge_accum<<<(E * C128 + 255) / 256, 256, 0, stream>>>(dst, eattr, loopA, deg, E);
  k_loop_fin<<<((int)NC + 255) / 256, 256, 0, stream>>>(loopA, deg, (int)NC);

  auto conv = [&](const float* xin, float* hout,
                  const float* Wl, const float* Wr, const float* We,
                  const float* att, const float* bias,
                  const float* bg, const float* bb, const float* bm, const float* bv) {
    gemm(xin, Wl, nullptr, xl, N, 0);
    gemm(xin, Wr, nullptr, xr, N, 0);
    gemm(eattr, We, nullptr, el_e, E, 0);
    gemm(loopA, We, nullptr, el_n, N, 0);
    k_fill<<<(N + 255) / 256, 256, 0, stream>>>(maxb, NEG_INF, N);
    hipMemsetAsync(sumb, 0, (size_t)N * sizeof(float), stream);
    int nw = E + N;
    k_logits<<<(nw + 7) / 8, 256, 0, stream>>>(src, dst, xl, xr, el_e, el_n, att,
                                               logits, maxb, E, N);
    k_expsum<<<(nw + 255) / 256, 256, 0, stream>>>(dst, logits, maxb, expb, sumb, E, N);
    hipMemsetAsync(hout, 0, NC * sizeof(float), stream);
    k_scatter<<<(nw * C128 + 255) / 256, 256, 0, stream>>>(src, dst, xl, expb, sumb,
                                                           hout, E, N);
    k_bias_bn_tanh<<<((int)NC + 255) / 256, 256, 0, stream>>>(hout, bias, bg, bb, bm,
                                                              bv, (int)NC);
  };

  conv(x,  h1, c1Wl, c1Wr, c1We, c1at, c1bi, bn1g, bn1b, bn1m, bn1v);
  conv(h1, h2, c2Wl, c2Wr, c2We, c2at, c2bi, bn2g, bn2b, bn2m, bn2v);

  // Attentional aggregation (xl/xr are free now; reuse as t/gate)
  float* tbuf = xl;
  float* gate = xr;
  gemm(h2, agW1, agb1, tbuf, N, 2);   // tanh(h@W1 + b1)
  gemm(tbuf, agW2, agb2, gate, N, 1); // t@W2 + b2
  k_fill<<<(G * C128 + 255) / 256, 256, 0, stream>>>(gmax, NEG_INF, G * C128);
  hipMemsetAsync(gsum,   0, (size_t)G * C128 * sizeof(float), stream);
  hipMemsetAsync(pooled, 0, (size_t)G * C128 * sizeof(float), stream);
  k_seg_max<<<((int)NC + 255) / 256, 256, 0, stream>>>(gate, batch, gmax, (int)NC);
  k_seg_exp<<<((int)NC + 255) / 256, 256, 0, stream>>>(gate, batch, gmax, gsum, (int)NC);
  k_seg_pool<<<((int)NC + 255) / 256, 256, 0, stream>>>(gate, batch, gsum, h2, pooled, (int)NC);

  float* out = (float*)d_out;
  // graph embedding head
  gemm(pooled, liW, lib, tmp1, G, 1);
  k_l2norm<<<(G + 7) / 8, 256, 0, stream>>>(tmp1, out, G);
  // image head
  gemm(imgf, imW, imb, tmp2, G, 1);
  k_bn_tanh<<<(G * C128 + 255) / 256, 256, 0, stream>>>(tmp2, big, bib, bim, biv, G * C128);
  k_l2norm<<<(G + 7) / 8, 256, 0, stream>>>(tmp2, out + (size_t)G * C128, G);
  // text head
  gemm(txtf, txW, txb, tmp3, G, 1);
  k_bn_tanh<<<(G * C128 + 255) / 256, 256, 0, stream>>>(tmp3, btg, btb, btm, btv, G * C128);
  k_l2norm<<<(G + 7) / 8, 256, 0, stream>>>(tmp3, out + 2 * (size_t)G * C128, G);
}